// DLRM_1082331758804
// MI455X (gfx1250) — compile-verified
//
#include <hip/hip_runtime.h>
#include <hip/hip_bf16.h>
#include <math.h>

typedef __attribute__((ext_vector_type(2))) float v2f;
typedef __attribute__((ext_vector_type(4))) float v4f;
typedef __attribute__((ext_vector_type(8))) float v8f;

#define BATCH     16384
#define NTAB      26
#define VOCAB_SZ  100000
#define DIM       64
#define NF        27      // 1 dense row + 26 embedding rows
#define INTER_COLS 442
#define INTER_LD   448    // padded row stride (multiple of 32)

static __device__ __forceinline__ v8f wmma_f32(v2f a, v2f b, v8f c) {
    // D = A(16x4 f32) * B(4x16 f32) + C(16x16 f32)
    return __builtin_amdgcn_wmma_f32_16x16x4_f32(
        /*neg_a=*/false, a, /*neg_b=*/false, b,
        /*c_mod=*/(short)0, c, /*reuse_a=*/false, /*reuse_b=*/false);
}

// ---------------------------------------------------------------------------
// C(M x N) = relu(A(M x K, row stride lda) @ W(K x N) + bias)
// Block tile 128x64, 8 waves arranged 4(M) x 2(N); each wave: 32x32 output.
// Software-pipelined: K-tile k+1 is fetched to registers while tile k computes.
// ---------------------------------------------------------------------------
__global__ __launch_bounds__(256)
void gemm_bias_relu(const float* __restrict__ A, int lda,
                    const float* __restrict__ W,
                    const float* __restrict__ bias,
                    float* __restrict__ C,
                    int N, int K)
{
    constexpr int BK = 32;
    __shared__ float As[128][BK + 4];   // stride 36: 16B-aligned rows, conflict-free
    __shared__ float Wt[64][BK + 2];    // W tile transposed: Wt[n][k], stride 34

    const int m0   = blockIdx.x * 128;
    const int n0   = blockIdx.y * 64;
    const int tid  = threadIdx.x;
    const int lane = tid & 31;
    const int wave = tid >> 5;
    const int rb   = (wave >> 1) * 32;   // wave row base in block tile
    const int cb   = (wave & 1) * 32;    // wave col base in block tile
    const int mr   = lane & 15;          // fragment row/col within 16
    const int ks   = (lane >> 4) * 2;    // K pair select (lanes 16-31 -> K=2,3)

    v8f c00 = {}, c01 = {}, c10 = {}, c11 = {};

    // cooperative-load mappings
    const int ar  = tid >> 1;            // A: 0..127 row
    const int ac  = (tid & 1) * 16;      // A: col half
    const int wk  = tid >> 3;            // W: 0..31 k
    const int wn8 = (tid & 7) * 8;       // W: 8-col group

    v4f regsA[4];                        // staged A fragment (16 floats)
    v4f regsW[2];                        // staged W fragment (8 floats)

    auto loadA = [&](int k0) {
        if (((lda & 3) == 0) && (k0 + BK <= lda)) {
            const float* src = A + (size_t)(m0 + ar) * lda + k0 + ac;
            regsA[0] = *(const v4f*)(src + 0);
            regsA[1] = *(const v4f*)(src + 4);
            regsA[2] = *(const v4f*)(src + 8);
            regsA[3] = *(const v4f*)(src + 12);
        } else {
            const float* srcRow = A + (size_t)(m0 + ar) * lda;
            #pragma unroll
            for (int j = 0; j < 16; ++j) {
                int k = k0 + ac + j;
                regsA[j >> 2][j & 3] = (k < K) ? srcRow[k] : 0.0f;
            }
        }
    };
    auto loadW = [&](int k0) {
        int kk = k0 + wk;
        if (kk < K) {
            const float* src = W + (size_t)kk * N + n0 + wn8;
            regsW[0] = *(const v4f*)(src + 0);
            regsW[1] = *(const v4f*)(src + 4);
        } else {
            regsW[0] = (v4f){0.f, 0.f, 0.f, 0.f};
            regsW[1] = (v4f){0.f, 0.f, 0.f, 0.f};
        }
    };

    loadA(0);
    loadW(0);

    for (int k0 = 0; k0 < K; k0 += BK) {
        __syncthreads();                 // previous tile fully consumed
        // ---- commit staged registers to LDS ----
        *(v4f*)&As[ar][ac + 0]  = regsA[0];
        *(v4f*)&As[ar][ac + 4]  = regsA[1];
        *(v4f*)&As[ar][ac + 8]  = regsA[2];
        *(v4f*)&As[ar][ac + 12] = regsA[3];
        #pragma unroll
        for (int j = 0; j < 4; ++j) Wt[wn8 + j][wk]     = regsW[0][j];
        #pragma unroll
        for (int j = 0; j < 4; ++j) Wt[wn8 + 4 + j][wk] = regsW[1][j];
        __syncthreads();

        // ---- prefetch next K-tile into registers (overlaps with WMMAs) ----
        int kn = k0 + BK;
        if (kn < K) {
            loadA(kn);
            loadW(kn);
            // warm L2 for tile k+2 (global_prefetch_b8)
            if (kn + BK < K) {
                __builtin_prefetch(A + (size_t)(m0 + ar) * lda + kn + BK + ac, 0, 1);
                __builtin_prefetch(W + (size_t)(kn + BK + wk) * N + n0 + wn8, 0, 1);
            }
        }

        // ---- 8 WMMA K-steps of 4 ----
        #pragma unroll
        for (int kk = 0; kk < BK; kk += 4) {
            v2f a0 = *(const v2f*)&As[rb + mr][kk + ks];
            v2f a1 = *(const v2f*)&As[rb + 16 + mr][kk + ks];
            v2f b0 = *(const v2f*)&Wt[cb + mr][kk + ks];
            v2f b1 = *(const v2f*)&Wt[cb + 16 + mr][kk + ks];
            c00 = wmma_f32(a0, b0, c00);
            c01 = wmma_f32(a0, b1, c01);
            c10 = wmma_f32(a1, b0, c10);
            c11 = wmma_f32(a1, b1, c11);
        }
    }

    // ---- epilogue: bias + relu + store ----
    const int rowHi = (lane >> 4) * 8;   // C/D layout: lanes 16-31 hold rows +8
    const int col0  = n0 + cb + mr;
    const int col1  = n0 + cb + 16 + mr;
    const float bv0 = bias[col0];
    const float bv1 = bias[col1];

    auto store_tile = [&](v8f acc, int mbase, int col, float bv) {
        #pragma unroll
        for (int v = 0; v < 8; ++v) {
            int row = m0 + mbase + rowHi + v;
            float val = acc[v] + bv;
            val = val > 0.0f ? val : 0.0f;
            C[(size_t)row * N + col] = val;
        }
    };
    store_tile(c00, rb,      col0, bv0);
    store_tile(c01, rb,      col1, bv1);
    store_tile(c10, rb + 16, col0, bv0);
    store_tile(c11, rb + 16, col1, bv1);
}

// ---------------------------------------------------------------------------
// Interaction: gather embeddings, build T (27x64 padded to 32), Z = T*T^T via
// WMMA (symmetric: only tiles (0,0),(0,1),(1,1)), scatter upper triangle.
// One wave per batch row; 4 waves per block.
// ---------------------------------------------------------------------------
__global__ __launch_bounds__(128)
void interaction_kernel(const float* __restrict__ dx,
                        const int*   __restrict__ sparse,
                        const float* __restrict__ tables,
                        float* __restrict__ inter)
{
    constexpr int TS = DIM + 4;          // stride 68: conflict-free fragment loads
    __shared__ float T[4][32][TS];
    const int lane = threadIdx.x & 31;
    const int w    = threadIdx.x >> 5;
    const int b    = blockIdx.x * 4 + w;
    const int mr   = lane & 15;
    const int ks   = (lane >> 4) * 2;
    const int c2   = lane * 2;           // this lane's 2 columns of a 64-wide row

    // row 0: bottom-MLP output; also copy into interaction cols [0,64)
    v2f d = *(const v2f*)(dx + (size_t)b * DIM + c2);
    *(v2f*)&T[w][0][c2] = d;
    *(v2f*)(inter + (size_t)b * INTER_LD + c2) = d;
    if (lane < INTER_LD - INTER_COLS)    // zero the 6 pad columns
        inter[(size_t)b * INTER_LD + INTER_COLS + lane] = 0.0f;

    // rows 1..26: embedding gathers (one full 256B row per wave per load)
    for (int t = 0; t < NTAB; ++t) {
        int idx = sparse[b * NTAB + t];
        const float* src = tables + ((size_t)t * VOCAB_SZ + (size_t)idx) * DIM;
        v2f e = *(const v2f*)(src + c2);
        *(v2f*)&T[w][1 + t][c2] = e;
    }
    // rows 27..31: zero padding
    #pragma unroll
    for (int r = NF; r < 32; ++r) {
        T[w][r][c2]     = 0.0f;
        T[w][r][c2 + 1] = 0.0f;
    }

    // Z = T * T^T : for a symmetric product the A-fragment of row-tile i and the
    // B-fragment of col-tile i are the same float2 loads from T.
    v8f z00 = {}, z01 = {}, z11 = {};
    #pragma unroll
    for (int kk = 0; kk < DIM; kk += 4) {
        v2f f0 = *(const v2f*)&T[w][mr][kk + ks];
        v2f f1 = *(const v2f*)&T[w][16 + mr][kk + ks];
        z00 = wmma_f32(f0, f0, z00);
        z01 = wmma_f32(f0, f1, z01);
        z11 = wmma_f32(f1, f1, z11);
    }

    // scatter row-major upper triangle (incl. diagonal) into cols [64, 442)
    float* outp = inter + (size_t)b * INTER_LD + DIM;
    const int rowHi = (lane >> 4) * 8;
    auto emit = [&](v8f acc, int ti, int tj) {
        int n = tj * 16 + mr;
        #pragma unroll
        for (int v = 0; v < 8; ++v) {
            int m = ti * 16 + rowHi + v;
            if (m < NF && n < NF && m <= n) {
                int pos = m * NF - (m * (m - 1)) / 2 + (n - m);
                outp[pos] = acc[v];
            }
        }
    };
    emit(z00, 0, 0);
    emit(z01, 0, 1);
    emit(z11, 1, 1);
}

// ---------------------------------------------------------------------------
// Final layer: out[b] = sigmoid(dot(X[b, 0:256], w) + bias)
// ---------------------------------------------------------------------------
__global__ __launch_bounds__(256)
void top_out_kernel(const float* __restrict__ X, const float* __restrict__ w,
                    const float* __restrict__ bias, float* __restrict__ out)
{
    __shared__ float wv[256];
    wv[threadIdx.x] = w[threadIdx.x];
    __syncthreads();
    const int b = blockIdx.x * 256 + threadIdx.x;
    const float* row = X + (size_t)b * 256;
    float s = 0.0f;
    #pragma unroll 8
    for (int k = 0; k < 256; k += 4) {
        v4f x = *(const v4f*)(row + k);
        s += x.x * wv[k] + x.y * wv[k + 1] + x.z * wv[k + 2] + x.w * wv[k + 3];
    }
    s += bias[0];
    out[b] = 1.0f / (1.0f + expf(-s));
}

// ---------------------------------------------------------------------------
extern "C" void kernel_launch(void* const* d_in, const int* in_sizes, int n_in,
                              void* d_out, int out_size, void* d_ws, size_t ws_size,
                              hipStream_t stream) {
    (void)in_sizes; (void)n_in; (void)out_size; (void)ws_size;
    const float* dense  = (const float*)d_in[0];
    const int*   sparse = (const int*)  d_in[1];
    const float* tables = (const float*)d_in[2];
    const float* bw1 = (const float*)d_in[3];
    const float* bb1 = (const float*)d_in[4];
    const float* bw2 = (const float*)d_in[5];
    const float* bb2 = (const float*)d_in[6];
    const float* bw3 = (const float*)d_in[7];
    const float* bb3 = (const float*)d_in[8];
    const float* tw1 = (const float*)d_in[9];
    const float* tb1 = (const float*)d_in[10];
    const float* tw2 = (const float*)d_in[11];
    const float* tb2 = (const float*)d_in[12];
    const float* tw3 = (const float*)d_in[13];
    const float* tb3 = (const float*)d_in[14];
    float* out = (float*)d_out;

    float* ws    = (float*)d_ws;
    float* h1    = ws;                                 // B*512
    float* h2    = h1 + (size_t)BATCH * 512;           // B*256
    float* dx    = h2 + (size_t)BATCH * 256;           // B*64
    float* inter = dx + (size_t)BATCH * 64;            // B*448 (zero-padded)
    float* t1 = h1;   // reuse: h1 dead after bottom L2
    float* t2 = h2;   // reuse: h2 dead after bottom L3

    dim3 blk(256);
    // bottom MLP: 13 -> 512 -> 256 -> 64, relu each
    gemm_bias_relu<<<dim3(BATCH / 128, 512 / 64), blk, 0, stream>>>(dense, 13,  bw1, bb1, h1, 512, 13);
    gemm_bias_relu<<<dim3(BATCH / 128, 256 / 64), blk, 0, stream>>>(h1,    512, bw2, bb2, h2, 256, 512);
    gemm_bias_relu<<<dim3(BATCH / 128,  64 / 64), blk, 0, stream>>>(h2,    256, bw3, bb3, dx,  64, 256);
    // embedding gather + pairwise interaction (self-GEMM upper triangle)
    interaction_kernel<<<dim3(BATCH / 4), dim3(128), 0, stream>>>(dx, sparse, tables, inter);
    // top MLP: 442 -> 512 -> 256 -> 1 (sigmoid in final kernel)
    gemm_bias_relu<<<dim3(BATCH / 128, 512 / 64), blk, 0, stream>>>(inter, INTER_LD, tw1, tb1, t1, 512, 442);
    gemm_bias_relu<<<dim3(BATCH / 128, 256 / 64), blk, 0, stream>>>(t1,    512,      tw2, tb2, t2, 256, 512);
    top_out_kernel<<<dim3(BATCH / 256), dim3(256), 0, stream>>>(t2, tw3, tb3, out);
}